// TGCNPO_82695300317854
// MI455X (gfx1250) — compile-verified
//
#include <hip/hip_runtime.h>
#include <hip/hip_bf16.h>
#include <math.h>

// ---------------------------------------------------------------------------
// TGCN forward, H0 = 0  =>  algebraic simplification:
//   agg computed once; R-gate dead; Z = sig(agg@Mz + cz), Ht = tanh(agg@Mh + ch)
//   H = (1-Z)*Ht ; out = softmax(relu(H) @ Wo + bo)
// Mz/Mh are precomputed *in WMMA B-fragment order* so the hot loop does one
// coalesced b64 load per v_wmma_f32_16x16x4_f32 operand.
// ---------------------------------------------------------------------------

typedef __attribute__((ext_vector_type(2))) float v2f;
typedef __attribute__((ext_vector_type(8))) float v8f;

#define IN_CH 64
#define OUT_CH 256
#define OSZ 4

// ---------------- degree ----------------
__global__ void tgcn_deg_kernel(const int* __restrict__ ei, float* __restrict__ deg, int E) {
    int e = blockIdx.x * blockDim.x + threadIdx.x;
    if (e < E) atomicAdd(&deg[ei[E + e]], 1.0f);
}

__global__ void tgcn_dinv_kernel(float* __restrict__ d, int N) {
    int i = blockIdx.x * blockDim.x + threadIdx.x;
    if (i < N) d[i] = rsqrtf(d[i] + 1.0f);
}

// ---------------- edge scatter: 16 threads per edge, float4 per thread -----
__global__ void tgcn_scatter_kernel(const float* __restrict__ x, const int* __restrict__ ei,
                                    const float* __restrict__ dinv, float* __restrict__ agg, int E) {
    long long t = (long long)blockIdx.x * blockDim.x + threadIdx.x;
    int e = (int)(t >> 4);
    if (e >= E) return;
    int part = (int)(t & 15);
    int s = ei[e];
    int d = ei[E + e];
    float w = dinv[s] * dinv[d];
    const float4 xv = *(const float4*)(x + (size_t)s * IN_CH + part * 4);
    float* ap = agg + (size_t)d * IN_CH + part * 4;
    atomicAdd(ap + 0, xv.x * w);
    atomicAdd(ap + 1, xv.y * w);
    atomicAdd(ap + 2, xv.z * w);
    atomicAdd(ap + 3, xv.w * w);
}

// ---------------- weight fusion into WMMA B-fragment layout ----------------
// Mf[((c*16 + kk)*32 + lane)*2 + t] = M[kk*4 + (lane>>4)*2 + t][c*16 + (lane&15)]
// where M = W @ Ltop (64x256). Row 64 of the grid computes cb = b@Ltop + Lb.
__global__ void tgcn_fuse_kernel(const float* __restrict__ W,   // [64,256]
                                 const float* __restrict__ b,   // [256]
                                 const float* __restrict__ L,   // [512,256] (top 256 rows used)
                                 const float* __restrict__ Lb,  // [256]
                                 float* __restrict__ Mf,        // [64*256] fragment-packed
                                 float* __restrict__ cb) {      // [256]
    int j = threadIdx.x;   // 0..255 : output column
    int i = blockIdx.x;    // 0..64  : K row; row 64 computes the fused bias
    if (i < IN_CH) {
        float s = 0.f;
        for (int k = 0; k < OUT_CH; ++k)
            s = fmaf(W[i * OUT_CH + k], L[(size_t)k * OUT_CH + j], s);
        // fragment-packed scatter store
        int kk = i >> 2, rem = i & 3;
        int halfk = rem >> 1, t = rem & 1;
        int c = j >> 4, lane16 = j & 15;
        int lane = halfk * 16 + lane16;
        Mf[((size_t)(c * 16 + kk) * 32 + lane) * 2 + t] = s;
    } else {
        float s = Lb[j];
        for (int k = 0; k < OUT_CH; ++k)
            s = fmaf(b[k], L[(size_t)k * OUT_CH + j], s);
        cb[j] = s;
    }
}

// ---------------- fused gates + head (WMMA f32 16x16x4) --------------------
// 8 waves / WG, 16 nodes / wave -> 128 nodes per workgroup.
__global__ __launch_bounds__(256) void tgcn_gates_head_kernel(
    const float* __restrict__ agg, const float* __restrict__ x,
    const float* __restrict__ dinv,
    const float* __restrict__ Mzf, const float* __restrict__ Mhf,
    const float* __restrict__ czb, const float* __restrict__ chb,
    const float* __restrict__ Wo, const float* __restrict__ bo,
    float* __restrict__ out, int N) {

    const int lane   = threadIdx.x & 31;
    const int wave   = threadIdx.x >> 5;
    const int lane16 = lane & 15;
    const int halfk  = lane >> 4;           // 0 | 1
    const int nodeBase = (blockIdx.x * 8 + wave) * 16;

    // ---- load 16x64 A tile (agg + self-loop term), clamped for tail -------
    int row = nodeBase + lane16;
    int rc = row < N ? row : N - 1;
    float dv = dinv[rc];
    float d2 = dv * dv;
    const size_t rbase = (size_t)rc * IN_CH;

    v2f afrag[16];
#pragma unroll
    for (int kk = 0; kk < 16; ++kk) {
        int k = kk * 4 + halfk * 2;
        v2f av = *(const v2f*)(agg + rbase + k);
        v2f xv = *(const v2f*)(x + rbase + k);
        v2f a;
        a.x = fmaf(xv.x, d2, av.x);
        a.y = fmaf(xv.y, d2, av.y);
        afrag[kk] = a;
    }

    float oacc[8][4];
#pragma unroll
    for (int v = 0; v < 8; ++v)
#pragma unroll
        for (int j = 0; j < 4; ++j) oacc[v][j] = 0.f;

    // ---- 16 column tiles of 16; both gates accumulated per tile -----------
    for (int c = 0; c < 16; ++c) {
        const int c0 = c * 16;
        const int colb = c0 + lane16;
        // fragment-packed bases: all 16 kk fragments within 4KB of these
        const float* pz = Mzf + ((size_t)c * 16 * 32 + lane) * 2;
        const float* ph = Mhf + ((size_t)c * 16 * 32 + lane) * 2;
        v8f accz = {};
        v8f acch = {};
#pragma unroll
        for (int kk = 0; kk < 16; ++kk) {
            v2f bz = *(const v2f*)(pz + kk * 64);
            v2f bh = *(const v2f*)(ph + kk * 64);
            accz = __builtin_amdgcn_wmma_f32_16x16x4_f32(
                false, afrag[kk], false, bz, (short)0, accz, false, false);
            acch = __builtin_amdgcn_wmma_f32_16x16x4_f32(
                false, afrag[kk], false, bh, (short)0, acch, false, false);
        }
        const float bzv = czb[colb];
        const float bhv = chb[colb];
        const float4 wrow = *(const float4*)(Wo + (size_t)colb * OSZ);
#pragma unroll
        for (int v = 0; v < 8; ++v) {
            float zz = 1.0f / (1.0f + __expf(-(accz[v] + bzv)));
            float th = tanhf(acch[v] + bhv);
            float h = (1.0f - zz) * th;           // H = (1-Z)*Ht  (Z*H0 = 0)
            h = h > 0.f ? h : 0.f;                // relu
            oacc[v][0] = fmaf(h, wrow.x, oacc[v][0]);
            oacc[v][1] = fmaf(h, wrow.y, oacc[v][1]);
            oacc[v][2] = fmaf(h, wrow.z, oacc[v][2]);
            oacc[v][3] = fmaf(h, wrow.w, oacc[v][3]);
        }
    }

    // ---- reduce partial head dots across the 16 lanes of each half --------
#pragma unroll
    for (int v = 0; v < 8; ++v)
#pragma unroll
        for (int j = 0; j < 4; ++j) {
            float s = oacc[v][j];
            s += __shfl_xor(s, 1, 32);
            s += __shfl_xor(s, 2, 32);
            s += __shfl_xor(s, 4, 32);
            s += __shfl_xor(s, 8, 32);
            oacc[v][j] = s;
        }

    const float b0 = bo[0], b1 = bo[1], b2 = bo[2], b3 = bo[3];
    // low half lanes v==lane16 write rows v; high half write rows v+8
#pragma unroll
    for (int v = 0; v < 8; ++v) {
        if (lane16 == v) {
            int node = nodeBase + v + halfk * 8;
            if (node < N) {
                float o0 = oacc[v][0] + b0;
                float o1 = oacc[v][1] + b1;
                float o2 = oacc[v][2] + b2;
                float o3 = oacc[v][3] + b3;
                float m = fmaxf(fmaxf(o0, o1), fmaxf(o2, o3));
                float e0 = __expf(o0 - m), e1 = __expf(o1 - m);
                float e2 = __expf(o2 - m), e3 = __expf(o3 - m);
                float inv = 1.0f / (e0 + e1 + e2 + e3);
                float* op = out + (size_t)node * OSZ;
                op[0] = e0 * inv;
                op[1] = e1 * inv;
                op[2] = e2 * inv;
                op[3] = e3 * inv;
            }
        }
    }
}

// ---------------------------------------------------------------------------
extern "C" void kernel_launch(void* const* d_in, const int* in_sizes, int n_in,
                              void* d_out, int out_size, void* d_ws, size_t ws_size,
                              hipStream_t stream) {
    const float* x   = (const float*)d_in[0];
    const int*   ei  = (const int*)d_in[1];
    const float* Wz  = (const float*)d_in[2];
    const float* bz  = (const float*)d_in[3];
    // d_in[4], d_in[5] (Wr, br) and d_in[10], d_in[11] (Lrw, Lrb): dead (H0==0)
    const float* Wh  = (const float*)d_in[6];
    const float* bh  = (const float*)d_in[7];
    const float* Lzw = (const float*)d_in[8];
    const float* Lzb = (const float*)d_in[9];
    const float* Lhw = (const float*)d_in[12];
    const float* Lhb = (const float*)d_in[13];
    const float* Wo  = (const float*)d_in[14];
    const float* bo  = (const float*)d_in[15];
    float* out = (float*)d_out;

    const int N = in_sizes[0] / IN_CH;
    const int E = in_sizes[1] / 2;

    // workspace carve-up (256B aligned slices)
    char* ws = (char*)d_ws;
    size_t off = 0;
    auto carve = [&](size_t bytes) {
        char* p = ws + off;
        off = (off + bytes + 255) & ~(size_t)255;
        return p;
    };
    float* dinv = (float*)carve((size_t)N * sizeof(float));
    float* agg  = (float*)carve((size_t)N * IN_CH * sizeof(float));
    float* Mzf  = (float*)carve((size_t)IN_CH * OUT_CH * sizeof(float));
    float* Mhf  = (float*)carve((size_t)IN_CH * OUT_CH * sizeof(float));
    float* czb  = (float*)carve((size_t)OUT_CH * sizeof(float));
    float* chb  = (float*)carve((size_t)OUT_CH * sizeof(float));

    hipMemsetAsync(dinv, 0, (size_t)N * sizeof(float), stream);
    hipMemsetAsync(agg,  0, (size_t)N * IN_CH * sizeof(float), stream);

    tgcn_deg_kernel<<<(E + 255) / 256, 256, 0, stream>>>(ei, dinv, E);
    tgcn_dinv_kernel<<<(N + 255) / 256, 256, 0, stream>>>(dinv, N);

    long long scatterThreads = (long long)E * 16;
    tgcn_scatter_kernel<<<(unsigned)((scatterThreads + 255) / 256), 256, 0, stream>>>(
        x, ei, dinv, agg, E);

    tgcn_fuse_kernel<<<IN_CH + 1, OUT_CH, 0, stream>>>(Wz, bz, Lzw, Lzb, Mzf, czb);
    tgcn_fuse_kernel<<<IN_CH + 1, OUT_CH, 0, stream>>>(Wh, bh, Lhw, Lhb, Mhf, chb);

    tgcn_gates_head_kernel<<<(N + 127) / 128, 256, 0, stream>>>(
        agg, x, dinv, Mzf, Mhf, czb, chb, Wo, bo, out, N);
}